// LaminoModule_29841432772778
// MI455X (gfx1250) — compile-verified
//
#include <hip/hip_runtime.h>
#include <hip/hip_bf16.h>
#include <math.h>

// ---------------------------------------------------------------------------
// Laminography forward projector (tike Lamino.fwd, Fourier-slice NUFFT) for
// MI455X / gfx1250.
//   1) centered 3D FFT of 256^3 complex volume  (3 passes of 256-pt line FFTs)
//   2) trilinear gather on tilted planes (M*256*256 samples, L2-resident vol)
//   3) centered 2D IFFT per angle               (2 passes of 256-pt line FFTs)
//
// Each 256-point FFT line = four-step algorithm with N1=N2=16:
//   A[x1][x2] (x = 16*x1 + x2) ;  B = W16 * A ;  B *= w256^(k1*x2) ;  D = B * W16
//   X[k1 + 16*k2] = D[k1][k2]
// Both 16x16 complex matmuls run on V_WMMA_F32_16X16X4_F32 (full fp32 WMMA),
// 4 K-chunks per matmul, 4 real WMMAs per complex matmul -> 32 WMMA / line.
//
// LDS is SoA (separate re/im planes + pre-negated W16-imag table) so every
// WMMA operand is produced directly by one ds_load_b64 / ds_load_2addr_b32
// with no register repacking or negation on the WMMA dependency chain.
//
// Centered transforms: fftshift/ifftshift folded into (-1)^x input and
// (-1)^k output checkerboards (exact for even N; the (-1)^(N/2) factor is +1).
// ---------------------------------------------------------------------------

typedef float v2f __attribute__((ext_vector_type(2)));
typedef float v8f __attribute__((ext_vector_type(8)));

#define NN    256
#define NN2   65536      // 256*256
#define MANG  180
#define TWO_PI 6.28318530717958647692f

// flat addressing of element t of line `line` for the three pass geometries
//  MODE 0: contiguous lines          f = line*256 + t         (axis z / axis u)
//  MODE 1: middle axis               f = (line>>8)*65536 + t*256 + (line&255)
//  MODE 2: slowest axis              f = t*65536 + line
template <int MODE>
__device__ __forceinline__ size_t flat_addr(int line, int t) {
  if (MODE == 0) return (size_t)line * NN + (size_t)t;
  if (MODE == 1) return (size_t)(line >> 8) * NN2 + (size_t)t * NN + (size_t)(line & 255);
  return (size_t)t * NN2 + (size_t)line;
}

// cooperative-load mapping: 256 threads x 8 iters cover 8 lines x 256 elems
// with consecutive tids touching consecutive global addresses.
template <int MODE>
__device__ __forceinline__ void coop_map(int tid, int j, int& sub, int& t) {
  if (MODE == 0) { sub = j; t = tid; }            // whole contiguous line per j
  else           { sub = tid & 7; t = j * 32 + (tid >> 3); } // fast-dim groups of 8
}

#define WMMA_F32X4(A, B, C) \
  __builtin_amdgcn_wmma_f32_16x16x4_f32(false, (A), false, (B), (short)0, (C), false, false)

// row-padded 16x17 plane
#define PLN (16 * 17)

// One pass of 256-point FFTs along one axis.
//  DIR = +1 : forward (w = e^{-2pi i /N}),  DIR = -1 : inverse (w = e^{+2pi i /N})
//  IN_CB_FULL : multiply input by (-1)^(t + par(line))  (pass 1: (-1)^(x+y+z))
//  IN_CB_T    : multiply input by (-1)^t                (inverse passes)
//  output always multiplied by (-1)^k and by `scale`.
template <int MODE, bool IN_CB_FULL, bool IN_CB_T, int DIR>
__global__ __launch_bounds__(256) void fft256_pass(const float2* __restrict__ in,
                                                   float2* __restrict__ out,
                                                   float scale) {
  __shared__ float Lre[8][PLN];     // 8 lines, re plane, padded 16x17
  __shared__ float Lim[8][PLN];     // 8 lines, im plane
  __shared__ float w16re[PLN];      // W16[a][b] = w16^(a*b)
  __shared__ float w16im[PLN];
  __shared__ float w16imn[PLN];     // -Im(W16): pre-negated for complex matmul
  __shared__ float2 tw[256];        // tw[p] = w256^p

  const int tid = threadIdx.x;
  const float DIRF = (float)DIR;

  // ---- build twiddle tables -------------------------------------------------
  {
    float ang = -DIRF * (TWO_PI / 256.0f) * (float)tid;
    float s, c; __sincosf(ang, &s, &c);
    tw[tid] = make_float2(c, s);
  }
  for (int idx = tid; idx < PLN; idx += 256) {
    int a = idx / 17, b = idx % 17;
    int pb = (a * (b & 15)) & 15;
    float ang = -DIRF * (TWO_PI / 16.0f) * (float)pb;
    float s, c; __sincosf(ang, &s, &c);
    w16re[idx] = c; w16im[idx] = s; w16imn[idx] = -s;
  }

  // ---- cooperative load of 8 lines into LDS (SoA) --------------------------
  const int lineBase = blockIdx.x * 8;
#pragma unroll
  for (int j = 0; j < 8; ++j) {
    int sub, t; coop_map<MODE>(tid, j, sub, t);
    const int line = lineBase + sub;
    float2 v = in[flat_addr<MODE>(line, t)];
    float s = 1.0f;
    if (IN_CB_FULL) s = ((t + (line >> 8) + line) & 1) ? -1.0f : 1.0f;
    if (IN_CB_T)    s = (t & 1) ? -1.0f : 1.0f;
    const int a = (t >> 4) * 17 + (t & 15);   // [x1][x2], t = 16*x1 + x2
    Lre[sub][a] = v.x * s;
    Lim[sub][a] = v.y * s;
  }
  __syncthreads();

  // ---- per-wave 256-pt FFT of one line -------------------------------------
  const int lane = tid & 31;
  const int wv   = tid >> 5;
  const int Mloc = lane & 15;      // WMMA row (A) / col (B,C,D) for this lane
  const int hi   = lane >> 4;      // half-wave select
  const int K0   = hi * 2;         // K offset inside a 4-wide chunk
  float* lre = Lre[wv];
  float* lim = Lim[wv];

  // Stage 1: C = W16 * A   (A-operand = W16 columns, B-operand = data rows)
  v8f cre = {0.f,0.f,0.f,0.f,0.f,0.f,0.f,0.f};
  v8f cim = {0.f,0.f,0.f,0.f,0.f,0.f,0.f,0.f};
#pragma unroll
  for (int c = 0; c < 4; ++c) {
    const int q0 = c * 4 + K0;
    // W16 rows are LDS-contiguous -> one b64 per operand, no repacking
    const v2f are  = *(const v2f*)&w16re [Mloc * 17 + q0];
    const v2f aim  = *(const v2f*)&w16im [Mloc * 17 + q0];
    const v2f aimn = *(const v2f*)&w16imn[Mloc * 17 + q0];
    // data columns: stride-17 pair -> ds_load_2addr_b32 lands as v2f
    v2f bre = { lre[q0 * 17 + Mloc], lre[(q0 + 1) * 17 + Mloc] };
    v2f bim = { lim[q0 * 17 + Mloc], lim[(q0 + 1) * 17 + Mloc] };
    cre = WMMA_F32X4(are,  bre, cre);
    cre = WMMA_F32X4(aimn, bim, cre);
    cim = WMMA_F32X4(are,  bim, cim);
    cim = WMMA_F32X4(aim,  bre, cim);
  }

  // Twiddle w256^(k1*x2) and stash B' back into this wave's LDS tile.
#pragma unroll
  for (int r = 0; r < 8; ++r) {
    const int k1 = r + 8 * hi;       // C/D layout: VGPR r -> row r + 8*hi
    const int x2 = Mloc;             // column = lane
    const float2 t = tw[(k1 * x2) & 255];
    lre[k1 * 17 + x2] = cre[r] * t.x - cim[r] * t.y;
    lim[k1 * 17 + x2] = cre[r] * t.y + cim[r] * t.x;
  }

  // Stage 2: D = B' * W16  (A-operand = B' chunks, B-operand = W16 rows)
  v8f dre = {0.f,0.f,0.f,0.f,0.f,0.f,0.f,0.f};
  v8f dim_ = {0.f,0.f,0.f,0.f,0.f,0.f,0.f,0.f};
#pragma unroll
  for (int c = 0; c < 4; ++c) {
    const int q0 = c * 4 + K0;
    // data rows are LDS-contiguous -> one b64 per operand
    const v2f are = *(const v2f*)&lre[Mloc * 17 + q0];
    const v2f aim = *(const v2f*)&lim[Mloc * 17 + q0];
    // W16 columns: stride-17 pair
    v2f bre  = { w16re [q0 * 17 + Mloc], w16re [(q0 + 1) * 17 + Mloc] };
    v2f bim  = { w16im [q0 * 17 + Mloc], w16im [(q0 + 1) * 17 + Mloc] };
    v2f bimn = { w16imn[q0 * 17 + Mloc], w16imn[(q0 + 1) * 17 + Mloc] };
    dre  = WMMA_F32X4(are, bre,  dre);
    dre  = WMMA_F32X4(aim, bimn, dre);
    dim_ = WMMA_F32X4(are, bim,  dim_);
    dim_ = WMMA_F32X4(aim, bre,  dim_);
  }

  // Output: k = k1 + 16*k2 ; sign (-1)^k = (-1)^r ; store transposed into tile
  // at [k>>4][k&15] = [k2][k1] so the cooperative store mapping can reuse it.
#pragma unroll
  for (int r = 0; r < 8; ++r) {
    const float s = (r & 1) ? -scale : scale;
    const int a = Mloc * 17 + (r + 8 * hi);
    lre[a] = dre[r] * s;
    lim[a] = dim_[r] * s;
  }
  __syncthreads();

  // ---- cooperative store ----------------------------------------------------
#pragma unroll
  for (int j = 0; j < 8; ++j) {
    int sub, t; coop_map<MODE>(tid, j, sub, t);
    const int a = (t >> 4) * 17 + (t & 15);
    out[flat_addr<MODE>(lineBase + sub, t)] = make_float2(Lre[sub][a], Lim[sub][a]);
  }
}

// ---------------------------------------------------------------------------
// Trilinear gather of the centered spectrum on tilted planes.
// grid: (256, MANG) blocks x 256 threads; one thread per detector pixel.
// The 134 MB spectrum fits in the 192 MB L2, so the irregular gathers are
// largely L2-resident.
// ---------------------------------------------------------------------------
__global__ __launch_bounds__(256) void lamino_sample(const float2* __restrict__ F,
                                                     const float* __restrict__ theta,
                                                     const float* __restrict__ tilt,
                                                     float2* __restrict__ S) {
  const int m = blockIdx.y;
  const int p = blockIdx.x * blockDim.x + threadIdx.x;  // 0..65535
  const int i = p >> 8;        // row  -> kv
  const int j = p & 255;       // col  -> ku

  const float th = theta[m];
  const float tl = tilt[0];
  float st, ct, sp, cp;
  __sincosf(th, &st, &ct);
  __sincosf(tl, &sp, &cp);

  const float fu = (float)(j - 128) * (1.0f / 256.0f);
  const float fv = (float)(i - 128) * (1.0f / 256.0f);

  // k3 = fu * e_u + fv * e_v ;  e_u = (ct, st, 0) ; e_v = (-st*cp, ct*cp, sp)
  const float pos0 = (fu * ct + fv * (-st * cp)) * 256.0f + 128.0f;
  const float pos1 = (fu * st + fv * ( ct * cp)) * 256.0f + 128.0f;
  const float pos2 = (            fv *  sp     ) * 256.0f + 128.0f;

  const float f0f = floorf(pos0), f1f = floorf(pos1), f2f = floorf(pos2);
  const int i0 = (int)f0f, i1 = (int)f1f, i2 = (int)f2f;
  const float fx = pos0 - f0f, fy = pos1 - f1f, fz = pos2 - f2f;

  float accre = 0.0f, accim = 0.0f;
#pragma unroll
  for (int dx = 0; dx < 2; ++dx) {
    const int x = i0 + dx;
    const float wx = dx ? fx : (1.0f - fx);
    if (x < 0 || x >= NN) continue;
#pragma unroll
    for (int dy = 0; dy < 2; ++dy) {
      const int y = i1 + dy;
      const float wy = dy ? fy : (1.0f - fy);
      if (y < 0 || y >= NN) continue;
      const size_t baseXY = ((size_t)x * NN + (size_t)y) * NN;
#pragma unroll
      for (int dz = 0; dz < 2; ++dz) {
        const int z = i2 + dz;
        if (z < 0 || z >= NN) continue;
        const float w = wx * wy * (dz ? fz : (1.0f - fz));
        const float2 v = F[baseXY + (size_t)z];
        accre = fmaf(w, v.x, accre);
        accim = fmaf(w, v.y, accim);
      }
    }
  }
  S[(size_t)m * NN2 + (size_t)p] = make_float2(accre, accim);
}

// ---------------------------------------------------------------------------
extern "C" void kernel_launch(void* const* d_in, const int* in_sizes, int n_in,
                              void* d_out, int out_size, void* d_ws, size_t ws_size,
                              hipStream_t stream) {
  (void)in_sizes; (void)n_in; (void)out_size; (void)ws_size;

  const float2* W     = (const float2*)d_in[0];   // (256,256,256,2) f32 == interleaved c64
  const float*  theta = (const float*)d_in[1];    // (180,)
  const float*  tilt  = (const float*)d_in[2];    // scalar
  float2*       outc  = (float2*)d_out;           // (180,256,256,2) == interleaved c64

  float2* F = (float2*)d_ws;                      // 256^3 complex64 = 134 MB
  float2* S = F + (size_t)NN * NN2;               // 180*256^2 complex64 = 94 MB

  const dim3 blk(256);
  const int volBlocks = NN2 / 8;                  // 8192 (65536 lines / 8)
  const int prjBlocks = (MANG * NN) / 8;          // 5760 (46080 lines / 8)

  // centered 3D forward FFT (input checkerboard applied once, in pass 1)
  fft256_pass<0, true,  false, +1><<<volBlocks, blk, 0, stream>>>(W, F, 1.0f);   // axis z
  fft256_pass<1, false, false, +1><<<volBlocks, blk, 0, stream>>>(F, F, 1.0f);   // axis y
  fft256_pass<2, false, false, +1><<<volBlocks, blk, 0, stream>>>(F, F, 1.0f);   // axis x

  // plane sampling (NUFFT gather)
  lamino_sample<<<dim3(NN, MANG), blk, 0, stream>>>(F, theta, tilt, S);

  // centered 2D inverse FFT per angle (1/256 scale per axis)
  fft256_pass<0, false, true, -1><<<prjBlocks, blk, 0, stream>>>(S, S,    1.0f / 256.0f); // axis u
  fft256_pass<1, false, true, -1><<<prjBlocks, blk, 0, stream>>>(S, outc, 1.0f / 256.0f); // axis v
}